// MixtureOfExpert_48120813584585
// MI455X (gfx1250) — compile-verified
//
#include <hip/hip_runtime.h>
#include <math.h>

// ---------------------------------------------------------------------------
// Problem constants (from the reference)
// ---------------------------------------------------------------------------
#define HID   768
#define G     12
#define TOT   12
#define B_SZ  64
#define S_SZ  128
#define SCALE (12.0f / 11.0f)

typedef __attribute__((ext_vector_type(2))) float v2f;
typedef __attribute__((ext_vector_type(4))) float v4f;
typedef __attribute__((ext_vector_type(8))) float v8f;

// ---------------------------------------------------------------------------
// Kernel 1: pooled[b,d] = mean_s input[b,s,d]   (25 MB, read-once -> NT loads)
// ---------------------------------------------------------------------------
__global__ void moe_pool_kernel(const float* __restrict__ in, float* __restrict__ pooled)
{
    int idx = blockIdx.x * blockDim.x + threadIdx.x;   // 0 .. B*HID
    if (idx >= B_SZ * HID) return;
    int b = idx / HID;
    int d = idx - b * HID;
    const float* p = in + (size_t)b * S_SZ * HID + d;
    float s = 0.0f;
    #pragma unroll 8
    for (int t = 0; t < S_SZ; ++t)
        s += __builtin_nontemporal_load(p + (size_t)t * HID);
    pooled[idx] = s * (1.0f / (float)S_SZ);
}

// ---------------------------------------------------------------------------
// Kernel 2: gating MLP. One block, 4 waves. Each wave computes a 16x16 tile of
// pooled @ W1 with V_WMMA_F32_16X16X4_F32 (exact fp32 -> bit-compatible with
// the f32 reference GEMM). W1 is staged in LDS zero-padded to N=16 so the
// inner loop has NO divergent guards: global_load_b64 (A) + 2x ds_load_b32 (B)
// + v_wmma per K-step.
// ---------------------------------------------------------------------------
__global__ __launch_bounds__(128) void moe_gating_kernel(
    const float* __restrict__ pooled,
    const float* __restrict__ W1, const float* __restrict__ b1,
    const float* __restrict__ W2, const float* __restrict__ b2,
    const float* __restrict__ cmask,
    float* __restrict__ prob_out, float* __restrict__ w_out)
{
    __shared__ float w1s[HID * 16];    // [k][n], n padded 12 -> 16 with zeros
    __shared__ float h_sh[B_SZ * G];

    const int tid  = threadIdx.x;
    const int wave = tid >> 5;         // 0..3 -> M tile
    const int lane = tid & 31;
    const int half = lane >> 4;        // 0: K pair {0,1}   1: K pair {2,3}
    const int l    = lane & 15;
    const int m0   = wave * 16;

    // Cooperative, branch-free staging of W1 into padded LDS.
    for (int i = tid; i < HID * 16; i += 128) {
        int k = i >> 4;
        int n = i & 15;
        float v = W1[(size_t)k * G + (n < G ? n : 0)];   // unconditional load
        w1s[i] = (n < G) ? v : 0.0f;                     // v_cndmask, no exec games
    }
    __syncthreads();

    v8f c = {};
    // 768/4 = 192 x V_WMMA_F32_16X16X4_F32
    #pragma unroll 4
    for (int kb = 0; kb < HID; kb += 4) {
        const int k0 = kb + half * 2;
        // A: 16x4 tile of pooled; contiguous K pair -> single b64 load per lane
        v2f a = *(const v2f*)(pooled + (size_t)(m0 + l) * HID + k0);
        // B: 4x16 tile from padded LDS (zero columns for n >= 12)
        v2f bb;
        bb.x = w1s[k0 * 16 + l];
        bb.y = w1s[k0 * 16 + 16 + l];
        c = __builtin_amdgcn_wmma_f32_16x16x4_f32(
                /*neg_a=*/false, a, /*neg_b=*/false, bb,
                /*c_mod=*/(short)0, c, /*reuse_a=*/false, /*reuse_b=*/false);
    }

    // C layout: vgpr r, lane -> row = m0 + 8*half + r, col = l
    if (l < G) {
        #pragma unroll
        for (int r = 0; r < 8; ++r) {
            int m = m0 + half * 8 + r;
            float x = c[r] + b1[l];
            // exact (erf-based) gelu, matching approximate=False
            float g = 0.5f * x * (1.0f + erff(x * 0.70710678118654752440f));
            h_sh[m * G + l] = g;
        }
    }
    __syncthreads();

    // Phase 2: one thread per batch row. 12x12 GEMM + softmax + mask matvec.
    if (tid < B_SZ) {
        float h[G];
        #pragma unroll
        for (int k = 0; k < G; ++k) h[k] = h_sh[tid * G + k];

        float e[G];
        float mx = -3.402823466e38f;
        #pragma unroll
        for (int n = 0; n < G; ++n) {
            float s = b2[n];
            #pragma unroll
            for (int k = 0; k < G; ++k) s += h[k] * W2[k * G + n];
            e[n] = s;
            mx = fmaxf(mx, s);
        }
        float denom = 0.0f;
        #pragma unroll
        for (int n = 0; n < G; ++n) { e[n] = expf(e[n] - mx); denom += e[n]; }
        float inv = 1.0f / denom;

        float p[G];
        #pragma unroll
        for (int n = 0; n < G; ++n) {
            p[n] = e[n] * inv;
            prob_out[tid * G + n] = p[n];
        }
        // w[g] = SCALE * sum_i p[i] * comb_mask[i,g]
        #pragma unroll
        for (int g = 0; g < G; ++g) {
            float s = 0.0f;
            #pragma unroll
            for (int i = 0; i < G; ++i) s += p[i] * cmask[i * G + g];
            w_out[tid * G + g] = s * SCALE;
        }
    }
}

// ---------------------------------------------------------------------------
// Kernel 3: the bandwidth kernel. One block per (b,s) row; each of 192 threads
// owns one float4 of the 768-wide feature dim. Per group g: one 128-bit NT
// load of bhm, one 128-bit NT store for the pass-through output, one FMA into
// the accumulator. bhm is read from HBM exactly once; the 604 MB of streaming
// load+copy traffic is marked non-temporal so it does not thrash the 192 MB
// L2. ~0.65 GB total -> ~28 us at 23.3 TB/s.
// ---------------------------------------------------------------------------
__global__ __launch_bounds__(192) void moe_combine_kernel(
    const float* __restrict__ bhm,
    const float* __restrict__ w_all,
    float* __restrict__ out_pm,
    float* __restrict__ out_copy)
{
    const int row = blockIdx.x;          // b * S + s
    const int b   = row >> 7;            // / S_SZ
    __shared__ float w[G];
    if (threadIdx.x < G) w[threadIdx.x] = w_all[b * G + threadIdx.x];
    __syncthreads();

    const v4f* __restrict__ src  = (const v4f*)(bhm      + (size_t)row * TOT * HID);
    v4f*       __restrict__ dstc = (v4f*)      (out_copy + (size_t)row * TOT * HID);
    v4f*       __restrict__ dstp = (v4f*)      (out_pm   + (size_t)row * HID);

    const int t = threadIdx.x;           // 0..191 (= HID/4 float4 slots per g-row)
    v4f acc = {0.0f, 0.0f, 0.0f, 0.0f};

    #pragma unroll
    for (int g = 0; g < TOT; ++g) {
        v4f v = __builtin_nontemporal_load(src + g * (HID / 4) + t);
        __builtin_nontemporal_store(v, dstc + g * (HID / 4) + t);  // pass-through
        const float wg = w[g];
        acc.x = fmaf(wg, v.x, acc.x);
        acc.y = fmaf(wg, v.y, acc.y);
        acc.z = fmaf(wg, v.z, acc.z);
        acc.w = fmaf(wg, v.w, acc.w);
    }
    __builtin_nontemporal_store(acc, dstp + t);
}

// ---------------------------------------------------------------------------
// Launch
// ---------------------------------------------------------------------------
extern "C" void kernel_launch(void* const* d_in, const int* in_sizes, int n_in,
                              void* d_out, int out_size, void* d_ws, size_t ws_size,
                              hipStream_t stream)
{
    const float* in_seq = (const float*)d_in[0];   // [B,S,HID]
    const float* bhm    = (const float*)d_in[1];   // [B,S,TOT,HID]
    const float* W1     = (const float*)d_in[2];   // [HID,G]
    const float* b1     = (const float*)d_in[3];   // [G]
    const float* W2     = (const float*)d_in[4];   // [G,G]
    const float* b2     = (const float*)d_in[5];   // [G]
    const float* cmask  = (const float*)d_in[6];   // [12,G]
    // d_in[7] = evaluate (scalar) -> deterministic combine path (the one used)

    // workspace: pooled [B,HID] then wv [B,G]
    float* pooled = (float*)d_ws;
    float* wv     = pooled + (size_t)B_SZ * HID;

    // output layout: prob [B,G] | prob_matrix [B,S,HID] | bhm copy [B,S,TOT,HID]
    float* out_prob = (float*)d_out;
    float* out_pm   = out_prob + (size_t)B_SZ * G;
    float* out_copy = out_pm + (size_t)B_SZ * S_SZ * HID;

    // 1) mean pool over seq
    {
        int n = B_SZ * HID;
        moe_pool_kernel<<<(n + 255) / 256, 256, 0, stream>>>(in_seq, pooled);
    }
    // 2) gating MLP with fp32 WMMA (single block, 4 waves)
    moe_gating_kernel<<<1, 128, 0, stream>>>(pooled, W1, b1, W2, b2, cmask,
                                             out_prob, wv);
    // 3) fused weighted-combine + pass-through copy (bandwidth bound)
    moe_combine_kernel<<<B_SZ * S_SZ, 192, 0, stream>>>(bhm, wv, out_pm, out_copy);
}